// Detector_50714973831524
// MI455X (gfx1250) — compile-verified
//
#include <hip/hip_runtime.h>
#include <stdint.h>

#define N_BOX   4096
#define NWORDS  128        // 4096 / 32 bits
#define THRESH  0.4f

#if defined(__AMDGCN__)
  #if __has_builtin(__builtin_amdgcn_global_load_async_to_lds_b32)
    #define HAVE_ASYNC_LDS 1
  #else
    #define HAVE_ASYNC_LDS 0
  #endif
#else
  #define HAVE_ASYNC_LDS 0
#endif

#if HAVE_ASYNC_LDS
typedef __attribute__((address_space(1))) int* as1_int_ptr;
typedef __attribute__((address_space(3))) int* as3_int_ptr;
#endif

__device__ __forceinline__ void wait_asynccnt0() {
#if defined(__AMDGCN__)
  #if __has_builtin(__builtin_amdgcn_s_wait_asynccnt)
    __builtin_amdgcn_s_wait_asynccnt(0);
  #else
    asm volatile("s_wait_asynccnt 0" ::: "memory");
  #endif
#endif
}

// ---------------------------------------------------------------------------
// Kernel 1: stable descending rank by confidence + gather boxes[:,1:5] into
// sorted order. O(N^2) counting sort: N=4096 -> ~16.7M compares total,
// trivial on MI455X. Confidences staged into LDS; on gfx1250 the stage uses
// the CDNA5 async global->LDS path (GLOBAL_LOAD_ASYNC_TO_LDS_B32, ASYNCcnt):
// per-lane gather addresses (stride-5) feed per-lane LDS destinations.
// ---------------------------------------------------------------------------
__global__ __launch_bounds__(256)
void rank_gather_kernel(const float* __restrict__ out5,
                        float* __restrict__ sboxes) {
    __shared__ float sconf[N_BOX];
    const int tid = threadIdx.x;
    const int i   = blockIdx.x * blockDim.x + tid;

#if HAVE_ASYNC_LDS
    for (int j = tid; j < N_BOX; j += blockDim.x) {
        int* g = (int*)(const_cast<float*>(out5) + (size_t)j * 5);
        int* l = (int*)(sconf + j);
        __builtin_amdgcn_global_load_async_to_lds_b32(
            (as1_int_ptr)g, (as3_int_ptr)l, /*offset=*/0, /*cpol=*/0);
    }
    wait_asynccnt0();      // my wave's async LDS writes are complete
    __syncthreads();       // publish across all waves in the block
#else
    for (int j = tid; j < N_BOX; j += blockDim.x)
        sconf[j] = out5[j * 5];
    __syncthreads();
#endif

    const float ci = sconf[i];
    int r = 0;
#pragma unroll 8
    for (int j = 0; j < N_BOX; ++j) {
        const float cj = sconf[j];
        // descending conf, stable (ascending original index on ties)
        r += (cj > ci) || (cj == ci && j < i);
    }
    sboxes[r * 4 + 0] = out5[i * 5 + 1];  // start
    sboxes[r * 4 + 1] = out5[i * 5 + 2];  // end
    sboxes[r * 4 + 2] = out5[i * 5 + 3];  // peak
    sboxes[r * 4 + 3] = out5[i * 5 + 4];  // height
}

// ---------------------------------------------------------------------------
// Kernel 2: build suppress bitmask matrix M[i][w], bit b = (piou(i, 32w+b) > T).
// One uint32 word per thread; 4096 rows x 128 words. Matrix is symmetric and
// the diagonal bit is set (piou(i,i)=1) but harmless: the scan tests bit i of
// "removed" BEFORE OR-ing row i. Entire box array (64KB) is L2/L0 resident.
// ---------------------------------------------------------------------------
__global__ __launch_bounds__(256)
void mask_kernel(const float* __restrict__ sb, uint32_t* __restrict__ M) {
    const int tid = blockIdx.x * blockDim.x + threadIdx.x;
    const int i = tid >> 7;              // row
    const int w = tid & (NWORDS - 1);    // word within row

    const float4 bi = ((const float4*)sb)[i];
    const float si = bi.x, ei = bi.y, pi = bi.z, hi = bi.w;
    const float ai = (ei - si) * hi;

    const float4* __restrict__ col = (const float4*)sb + (w << 5);
    uint32_t bits = 0u;
#pragma unroll 4
    for (int b = 0; b < 32; ++b) {
        const float4 bj = col[b];
        const float sj = bj.x, ej = bj.y, pj = bj.z, hj = bj.w;

        float il    = fminf(ei, ej) - fmaxf(si, sj);
        il          = fmaxf(il, 0.0f);
        const float inter = il * fminf(hi, hj);
        const float aj    = (ej - sj) * hj;
        const float uni   = ai + aj - inter;
        const float iou   = inter / uni;               // uni > 0 always here
        const float pd    = fabsf(pi - pj);
        const float ud    = fabsf(fmaxf(ei, ej) - fminf(si, sj));
        const float piou  = iou - pd / ud;
        if (piou > THRESH) bits |= (1u << b);
    }
    M[tid] = bits;
}

// ---------------------------------------------------------------------------
// Kernel 3: sequential greedy scan, ONE wave32. The 4096-bit "removed" mask
// lives in 4 VGPRs per lane (lane l owns words l, l+32, l+64, l+96). Per box:
// broadcast the relevant removed word via __shfl (uniform select -> cheap),
// and only for kept boxes OR in the 512B mask row. global_prefetch_b8 pulls
// the next row toward the caches to hide latency on the sequential chain.
// Kept condition is wave-uniform -> scalar branch, no divergence.
// ---------------------------------------------------------------------------
__global__ __launch_bounds__(32)
void nms_scan_kernel(const uint32_t* __restrict__ M, uint32_t* __restrict__ keep) {
    const int lane = threadIdx.x;  // 0..31, wave32
    uint32_t r0 = 0u, r1 = 0u, r2 = 0u, r3 = 0u;

    for (int i = 0; i < N_BOX; ++i) {
        const uint32_t* __restrict__ row = M + ((size_t)i << 7);
        // prefetch next row (gfx1250 global_prefetch_b8); speculative, OOB-safe
        __builtin_prefetch((const void*)(row + NWORDS + lane * 4), 0, 3);

        const int wi  = i >> 5;        // which removed-word holds bit i
        const int sel = wi >> 5;       // which register (uniform)
        const int src = wi & 31;       // which lane (uniform)
        const uint32_t mine = (sel == 0) ? r0 : (sel == 1) ? r1
                            : (sel == 2) ? r2 : r3;
        const uint32_t v    = __shfl(mine, src, 32);
        const uint32_t kept = ((v >> (i & 31)) & 1u) ^ 1u;

        if (lane == 0) keep[i] = kept;
        if (kept) {
            r0 |= row[lane];
            r1 |= row[lane + 32];
            r2 |= row[lane + 64];
            r3 |= row[lane + 96];
        }
    }
}

// ---------------------------------------------------------------------------
// Kernel 4: emit outputs. d_out layout (tuple concat, float32):
//   [0 .. 4*N)      kept_boxes = keep ? (s,e,p,h) : 0
//   [4*N .. 5*N)    keep flags as 1.0 / 0.0
// ---------------------------------------------------------------------------
__global__ __launch_bounds__(256)
void write_out_kernel(const float* __restrict__ sb,
                      const uint32_t* __restrict__ keep,
                      float* __restrict__ out) {
    const int i = blockIdx.x * blockDim.x + threadIdx.x;
    const float k = keep[i] ? 1.0f : 0.0f;
    const float4 b = ((const float4*)sb)[i];
    float4 ob;
    ob.x = b.x * k; ob.y = b.y * k; ob.z = b.z * k; ob.w = b.w * k;
    ((float4*)out)[i] = ob;
    out[N_BOX * 4 + i] = k;
}

// ---------------------------------------------------------------------------
extern "C" void kernel_launch(void* const* d_in, const int* in_sizes, int n_in,
                              void* d_out, int out_size, void* d_ws, size_t ws_size,
                              hipStream_t stream) {
    (void)in_sizes; (void)n_in; (void)out_size; (void)ws_size;

    const float* out5 = (const float*)d_in[0];          // 4096 x 5 f32

    // workspace layout
    float*    sboxes = (float*)d_ws;                                  //  64 KB
    uint32_t* keep   = (uint32_t*)((char*)d_ws + 64 * 1024);          //  16 KB
    uint32_t* M      = (uint32_t*)((char*)d_ws + 128 * 1024);         //   2 MB

    rank_gather_kernel<<<N_BOX / 256, 256, 0, stream>>>(out5, sboxes);
    mask_kernel<<<(N_BOX * NWORDS) / 256, 256, 0, stream>>>(sboxes, M);
    nms_scan_kernel<<<1, 32, 0, stream>>>(M, keep);
    write_out_kernel<<<N_BOX / 256, 256, 0, stream>>>(sboxes, keep, (float*)d_out);
}